// MambaMNIST_24180665876799
// MI455X (gfx1250) — compile-verified
//
#include <hip/hip_runtime.h>
#include <hip/hip_bf16.h>
#include <math.h>

// Problem constants (match reference)
#define BATCH   4096
#define INPUTD  784
#define DIM     1024
#define NLAYERS 4
#define D_STATE 16
#define D_INNER 2048
#define DT_RANK 64
#define XP_N    (DT_RANK + 2 * D_STATE)   // 96
#define NCLS    10

typedef __attribute__((ext_vector_type(2))) float v2f;
typedef __attribute__((ext_vector_type(8))) float v8f;

// ---------------------------------------------------------------------------
// Tiled f32 WMMA GEMM:  C[r,c] = epi( sum_k A[r,k]*B[k,c] + bias[c] ) (+ Cin)
//   block tile 128x64, K-chunk 32, 256 threads (8 waves), each wave -> 16x64.
//   A in LDS as [m][k] (even stride 34 -> ds_load_b64 frags),
//   B in LDS transposed [n][k] (odd stride 33 -> adjacent-pair frags, no movs,
//   conflict-free stores and reads). Uses V_WMMA_F32_16X16X4_F32.
// ---------------------------------------------------------------------------
#define TM 128
#define TN 64
#define TK 32
#define LDA_S 34   // even: keeps A-frag pairs 8B aligned for b64 LDS loads
#define LDBT  33   // odd: conflict-free transposed stores + fragment reads

template <int SOFTPLUS, int ACCUM>
__global__ void __launch_bounds__(256)
gemm_wmma_f32(const float* __restrict__ A, int lda,
              const float* __restrict__ Bw, int ldb,
              const float* __restrict__ bias,
              const float* __restrict__ Cin,
              float* __restrict__ C, int ldc,
              int M, int N, int K)
{
    __shared__ float As[TM * LDA_S];     // 17408 B
    __shared__ float Bs[TN * LDBT];      //  8448 B

    const int tid  = threadIdx.x;
    const int wave = tid >> 5;          // 0..7
    const int lane = tid & 31;
    const int m0 = blockIdx.y * TM;
    const int n0 = blockIdx.x * TN;

    v8f acc0 = {}; v8f acc1 = {}; v8f acc2 = {}; v8f acc3 = {};

    // A-frag: lanes 0-15 -> K=k+0,k+1 ; lanes 16-31 -> K=k+2,k+3 (same 16 rows)
    const int kl = (lane >= 16) ? 2 : 0;
    const int am = wave * 16 + (lane & 15);
    const int bn = lane & 15;

    for (int k0 = 0; k0 < K; k0 += TK) {
        __syncthreads();
        // ---- load A tile (TM x TK), coalesced along K ----
        {
            const int kk = tid & 31;
            const int mb = (tid >> 5) * 16;
            const bool gk = (k0 + kk) < K;
            #pragma unroll
            for (int i = 0; i < 16; ++i) {
                const int m = mb + i;
                float v = 0.0f;
                if (gk && (m0 + m) < M)
                    v = A[(size_t)(m0 + m) * lda + (k0 + kk)];
                As[m * LDA_S + kk] = v;
            }
        }
        // ---- load B tile (TK x TN) transposed into LDS [n][k] ----
        {
            const int nn = tid & 63;
            const int kb = (tid >> 6) * 8;
            const bool gn = (n0 + nn) < N;
            #pragma unroll
            for (int i = 0; i < 8; ++i) {
                const int k = kb + i;
                float v = 0.0f;
                if (gn && (k0 + k) < K)
                    v = Bw[(size_t)(k0 + k) * ldb + (n0 + nn)];
                Bs[nn * LDBT + k] = v;
            }
        }
        __syncthreads();

        // Prefetch next K-chunk while this one computes (global_prefetch_b8).
        {
            const int pk = k0 + TK;
            if (pk < K) {
                const int pm = m0 + (tid >> 1);
                if (pm < M)
                    __builtin_prefetch(&A[(size_t)pm * lda + pk], 0, 3);
                const int pr = pk + (tid & 31);
                if (pr < K)
                    __builtin_prefetch(&Bw[(size_t)pr * ldb + n0], 0, 3);
            }
        }

        // ---- compute: 8 K-steps x 4 N-subtiles of v_wmma_f32_16x16x4_f32 ----
        #pragma unroll
        for (int kk = 0; kk < TK; kk += 4) {
            v2f a;
            {
                const float* ap = &As[am * LDA_S + kk + kl];
                a.x = ap[0]; a.y = ap[1];
            }
            v2f b0, b1, b2, b3;
            {
                const float* bp = &Bs[bn * LDBT + kk + kl];
                b0.x = bp[0];              b0.y = bp[1];
                b1.x = bp[16 * LDBT];      b1.y = bp[16 * LDBT + 1];
                b2.x = bp[32 * LDBT];      b2.y = bp[32 * LDBT + 1];
                b3.x = bp[48 * LDBT];      b3.y = bp[48 * LDBT + 1];
            }
            acc0 = __builtin_amdgcn_wmma_f32_16x16x4_f32(false, a, false, b0, (short)0, acc0, false, false);
            acc1 = __builtin_amdgcn_wmma_f32_16x16x4_f32(false, a, false, b1, (short)0, acc1, false, false);
            acc2 = __builtin_amdgcn_wmma_f32_16x16x4_f32(false, a, false, b2, (short)0, acc2, false, false);
            acc3 = __builtin_amdgcn_wmma_f32_16x16x4_f32(false, a, false, b3, (short)0, acc3, false, false);
        }
    }

    // ---- epilogue: D layout = VGPR v: lanes 0-15 -> M=v, lanes 16-31 -> M=v+8
    const int rbase = m0 + wave * 16 + ((lane >= 16) ? 8 : 0);
    const int cc = lane & 15;
    #pragma unroll
    for (int s = 0; s < 4; ++s) {
        const int c = n0 + s * 16 + cc;
        if (c >= N) continue;
        const float bv = bias ? bias[c] : 0.0f;
        v8f acc = (s == 0) ? acc0 : (s == 1) ? acc1 : (s == 2) ? acc2 : acc3;
        #pragma unroll
        for (int v = 0; v < 8; ++v) {
            const int r = rbase + v;
            if (r < M) {
                float val = acc[v] + bv;
                if (SOFTPLUS) val = (val > 20.0f) ? val : log1pf(expf(val));
                if (ACCUM)    val += Cin[(size_t)r * ldc + c];
                C[(size_t)r * ldc + c] = val;
            }
        }
    }
}

// ---------------------------------------------------------------------------
// RMSNorm: one wave32 per row of (BATCH x DIM)
// ---------------------------------------------------------------------------
__global__ void __launch_bounds__(256)
rmsnorm_kernel(const float* __restrict__ x, const float* __restrict__ w,
               float* __restrict__ out)
{
    const int row  = blockIdx.x * 8 + (threadIdx.x >> 5);
    const int lane = threadIdx.x & 31;
    const float* xr = x + (size_t)row * DIM;
    float ss = 0.0f;
    #pragma unroll
    for (int i = lane; i < DIM; i += 32) { const float v = xr[i]; ss += v * v; }
    #pragma unroll
    for (int off = 16; off > 0; off >>= 1) ss += __shfl_xor(ss, off, 32);
    const float scale = rsqrtf(ss * (1.0f / DIM) + 1e-6f);
    float* orow = out + (size_t)row * DIM;
    #pragma unroll
    for (int i = lane; i < DIM; i += 32) orow[i] = xr[i] * scale * w[i];
}

// T=1 causal depthwise conv collapses to last tap + bias, then SiLU
__global__ void __launch_bounds__(256)
convsilu_kernel(const float* __restrict__ xz, const float* __restrict__ convW,
                const float* __restrict__ convb, float* __restrict__ xc)
{
    const int idx = blockIdx.x * 256 + threadIdx.x;      // BATCH * D_INNER
    const int b = idx >> 11, d = idx & (D_INNER - 1);
    const float v = xz[(size_t)b * (2 * D_INNER) + d] * convW[d * 4 + 3] + convb[d];
    xc[idx] = v / (1.0f + expf(-v));
}

// bc[b] = sum_n Bm[b,n]*Cm[b,n]  (x_db cols 64..79 and 80..95)
__global__ void __launch_bounds__(256)
bc_kernel(const float* __restrict__ xdb, float* __restrict__ bc)
{
    const int b = blockIdx.x * 256 + threadIdx.x;
    const float* r = xdb + (size_t)b * XP_N;
    float s = 0.0f;
    #pragma unroll
    for (int n = 0; n < D_STATE; ++n) s += r[DT_RANK + n] * r[DT_RANK + D_STATE + n];
    bc[b] = s;
}

// y = (xc*dt*bc[b] + Dp*xc) * silu(z)
__global__ void __launch_bounds__(256)
ygate_kernel(const float* __restrict__ xc, const float* __restrict__ dtv,
             const float* __restrict__ bc, const float* __restrict__ Dp,
             const float* __restrict__ xz, float* __restrict__ y)
{
    const int idx = blockIdx.x * 256 + threadIdx.x;
    const int b = idx >> 11, d = idx & (D_INNER - 1);
    const float xcv = xc[idx];
    const float yv  = xcv * dtv[idx] * bc[b] + Dp[d] * xcv;
    const float z   = xz[(size_t)b * (2 * D_INNER) + D_INNER + d];
    y[idx] = yv * (z / (1.0f + expf(-z)));
}

// ---------------------------------------------------------------------------
extern "C" void kernel_launch(void* const* d_in, const int* in_sizes, int n_in,
                              void* d_out, int out_size, void* d_ws, size_t ws_size,
                              hipStream_t stream) {
    (void)in_sizes; (void)n_in; (void)out_size; (void)ws_size;
    const float* x       = (const float*)d_in[0];
    const float* Wi      = (const float*)d_in[1];
    const float* bi      = (const float*)d_in[2];
    const float* norm_w  = (const float*)d_in[3];
    const float* inW     = (const float*)d_in[4];
    const float* inb     = (const float*)d_in[5];
    const float* convW   = (const float*)d_in[6];
    const float* convb   = (const float*)d_in[7];
    const float* xpW     = (const float*)d_in[8];
    const float* dtW     = (const float*)d_in[9];
    const float* dtb     = (const float*)d_in[10];
    /* d_in[11] = A_log: provably unused at T=1 (state starts at zero) */
    const float* Dp      = (const float*)d_in[12];
    const float* outW    = (const float*)d_in[13];
    const float* outb    = (const float*)d_in[14];
    const float* normf_w = (const float*)d_in[15];
    const float* headW   = (const float*)d_in[16];
    const float* headb   = (const float*)d_in[17];
    float* logits = (float*)d_out;

    // scratch layout
    float* ws  = (float*)d_ws;
    float* h   = ws;                               // BATCH*DIM
    float* hn  = h   + (size_t)BATCH * DIM;        // BATCH*DIM
    float* xz  = hn  + (size_t)BATCH * DIM;        // BATCH*2*D_INNER
    float* xc  = xz  + (size_t)BATCH * 2 * D_INNER;// BATCH*D_INNER
    float* xdb = xc  + (size_t)BATCH * D_INNER;    // BATCH*XP_N
    float* dtv = xdb + (size_t)BATCH * XP_N;       // BATCH*D_INNER
    float* bc  = dtv + (size_t)BATCH * D_INNER;    // BATCH
    float* yb  = bc  + (size_t)BATCH;              // BATCH*D_INNER

    const dim3 blk(256);
    auto grid = [](int M, int N) { return dim3((N + TN - 1) / TN, (M + TM - 1) / TM); };

    // h = x @ Wi + bi            (M=4096, K=784, N=1024)
    gemm_wmma_f32<0, 0><<<grid(BATCH, DIM), blk, 0, stream>>>(
        x, INPUTD, Wi, DIM, bi, nullptr, h, DIM, BATCH, DIM, INPUTD);

    for (int L = 0; L < NLAYERS; ++L) {
        const float* nw   = norm_w + (size_t)L * DIM;
        const float* inWl = inW    + (size_t)L * DIM * 2 * D_INNER;
        const float* inbl = inb    + (size_t)L * 2 * D_INNER;
        const float* cWl  = convW  + (size_t)L * D_INNER * 4;
        const float* cbl  = convb  + (size_t)L * D_INNER;
        const float* xpWl = xpW    + (size_t)L * D_INNER * XP_N;
        const float* dtWl = dtW    + (size_t)L * DT_RANK * D_INNER;
        const float* dtbl = dtb    + (size_t)L * D_INNER;
        const float* Dpl  = Dp     + (size_t)L * D_INNER;
        const float* oWl  = outW   + (size_t)L * D_INNER * DIM;
        const float* obl  = outb   + (size_t)L * DIM;

        // hn = rmsnorm(h) * norm_w
        rmsnorm_kernel<<<BATCH / 8, 256, 0, stream>>>(h, nw, hn);
        // xz = hn @ inW + inb     (K=1024, N=4096)
        gemm_wmma_f32<0, 0><<<grid(BATCH, 2 * D_INNER), blk, 0, stream>>>(
            hn, DIM, inWl, 2 * D_INNER, inbl, nullptr, xz, 2 * D_INNER,
            BATCH, 2 * D_INNER, DIM);
        // xc = silu(xpart * convW[:,3] + convb)
        convsilu_kernel<<<(BATCH * D_INNER) / 256, 256, 0, stream>>>(xz, cWl, cbl, xc);
        // xdb = xc @ xpW           (K=2048, N=96)
        gemm_wmma_f32<0, 0><<<grid(BATCH, XP_N), blk, 0, stream>>>(
            xc, D_INNER, xpWl, XP_N, nullptr, nullptr, xdb, XP_N,
            BATCH, XP_N, D_INNER);
        // bc[b] = B . C
        bc_kernel<<<BATCH / 256, 256, 0, stream>>>(xdb, bc);
        // dtv = softplus(dt_lr @ dtW + dtb)   (K=64, N=2048)
        gemm_wmma_f32<1, 0><<<grid(BATCH, D_INNER), blk, 0, stream>>>(
            xdb, XP_N, dtWl, D_INNER, dtbl, nullptr, dtv, D_INNER,
            BATCH, D_INNER, DT_RANK);
        // y = (xc*dt*bc + Dp*xc) * silu(z)
        ygate_kernel<<<(BATCH * D_INNER) / 256, 256, 0, stream>>>(xc, dtv, bc, Dpl, xz, yb);
        // h = h + y @ outW + outb  (K=2048, N=1024, residual accumulate)
        gemm_wmma_f32<0, 1><<<grid(BATCH, DIM), blk, 0, stream>>>(
            yb, D_INNER, oWl, DIM, obl, h, h, DIM, BATCH, DIM, D_INNER);
    }

    // final norm + head
    rmsnorm_kernel<<<BATCH / 8, 256, 0, stream>>>(h, normf_w, hn);
    gemm_wmma_f32<0, 0><<<grid(BATCH, NCLS), blk, 0, stream>>>(
        hn, DIM, headW, NCLS, headb, nullptr, logits, NCLS, BATCH, NCLS, DIM);
}